// SequentialLSTM_52819507806513
// MI455X (gfx1250) — compile-verified
//
#include <hip/hip_runtime.h>
#include <math.h>

// ---------------------------------------------------------------------------
// Shapes (fixed by the reference): B=4, S=4096, D=512, FH=2048, H=64
// ---------------------------------------------------------------------------
#define BB 4
#define SS 4096
#define DD 512
#define FHH 2048
#define HH 64
#define NT 8          // 16x16 N-tiles per wave (wave tile = 16M x 128N)

typedef __attribute__((ext_vector_type(2))) float v2f;
typedef __attribute__((ext_vector_type(8))) float v8f;
typedef __attribute__((ext_vector_type(4))) int   v4i;

#if __has_builtin(__builtin_amdgcn_global_load_async_to_lds_b128) && \
    __has_builtin(__builtin_amdgcn_s_wait_asynccnt)
#define HAVE_ASYNC_LDS 1
// Signature (from round-3/4 diagnostics): (AS1 v4i*, AS3 v4i*, imm offset, imm cpol)
typedef __attribute__((address_space(1))) v4i* gvec_t;
typedef __attribute__((address_space(3))) v4i* lvec_t;
#else
#define HAVE_ASYNC_LDS 0
#endif

// ---------------------------------------------------------------------------
// Kernel 1: Weff[i][j] = sum_k W2[i][k] * W1[k][j]   (512x512, K=2048)
// ---------------------------------------------------------------------------
__global__ __launch_bounds__(256)
void fuse_weights(const float* __restrict__ W1, const float* __restrict__ W2,
                  float* __restrict__ Weff) {
  int idx = blockIdx.x * 256 + threadIdx.x;      // 0 .. 512*512-1
  int i = idx >> 9;
  int j = idx & 511;
  const float* w2r = W2 + (size_t)i * FHH;
  float s = 0.f;
  #pragma unroll 4
  for (int k = 0; k < FHH; ++k)
    s = fmaf(w2r[k], W1[(size_t)k * DD + j], s);
  Weff[idx] = s;
}

// ---------------------------------------------------------------------------
// Kernel 2: beff[i] = W2[i,:]·b1 + b2[i]  (512);  bsum[g] = bih[g]+bhh[g] (256)
// ---------------------------------------------------------------------------
__global__ __launch_bounds__(512)
void prep_bias(const float* __restrict__ W2, const float* __restrict__ b1,
               const float* __restrict__ b2, const float* __restrict__ bih,
               const float* __restrict__ bhh,
               float* __restrict__ beff, float* __restrict__ bsum) {
  int i = threadIdx.x;
  if (i < 512) {
    const float* w2r = W2 + (size_t)i * FHH;
    float s = b2[i];
    #pragma unroll 4
    for (int k = 0; k < FHH; ++k) s = fmaf(w2r[k], b1[k], s);
    beff[i] = s;
  }
  if (i < 256) bsum[i] = bih[i] + bhh[i];
}

// ---------------------------------------------------------------------------
// Kernel 3: WMMA f32 GEMM:  C[m][n] = sum_k A[m][k]*W[n][k] + bias[n]
//   K is a template constant so every fragment load in the K-loop is an
//   immediate offset (t*16*K*4 bytes, max 224KB << 24-bit IOFFSET) off just
//   TWO base pointers -> minimal address VGPRs, pipeline regs stay live.
//   Block = 128 threads = 4 waves stacked in M; each wave: 16M x 128N strip,
//   NT=8 f32 accumulators, K-step 4 (V_WMMA_F32_16X16X4_F32), depth-1
//   register double-buffering.
// ---------------------------------------------------------------------------
template <int K>
__global__ __launch_bounds__(128)
void wmma_gemm_bias(const float* __restrict__ A, const float* __restrict__ W,
                    const float* __restrict__ bias, float* __restrict__ C,
                    int N) {
  const int lane = threadIdx.x & 31;
  const int wave = threadIdx.x >> 5;
  const int half = lane >> 4;       // 0 or 1
  const int r    = lane & 15;       // 0..15
  const int m0 = (blockIdx.x * 4 + wave) * 16;
  const int n0 = blockIdx.y * (16 * NT);

  // Two base pointers; everything else is immediate offsets.
  const float* ap = A + (size_t)(m0 + r) * K + 2 * half;
  const float* wp = W + (size_t)(n0 + r) * K + 2 * half;

  v8f acc[NT];
  #pragma unroll
  for (int t = 0; t < NT; ++t) acc[t] = (v8f){};

  // Pipeline prologue: fragments for K-chunk 0.
  v2f a_c = *(const v2f*)(ap);
  v2f b_c[NT];
  #pragma unroll
  for (int t = 0; t < NT; ++t) b_c[t] = *(const v2f*)(wp + t * 16 * K);

  for (int kk = 0; kk < K - 4; kk += 4) {
    // Stage K-chunk kk+4 while chunk kk computes (immediate-offset loads).
    v2f a_n = *(const v2f*)(ap + kk + 4);
    v2f b_n[NT];
    #pragma unroll
    for (int t = 0; t < NT; ++t)
      b_n[t] = *(const v2f*)(wp + t * 16 * K + kk + 4);
    // Stream the HBM-resident A operand ahead (global_prefetch_b8);
    // W is L2-resident (<=1MB vs 192MB L2).
    __builtin_prefetch(ap + kk + 68, 0, 1);

    #pragma unroll
    for (int t = 0; t < NT; ++t)
      acc[t] = __builtin_amdgcn_wmma_f32_16x16x4_f32(false, a_c, false, b_c[t],
                                                     (short)0, acc[t], false, false);
    a_c = a_n;
    #pragma unroll
    for (int t = 0; t < NT; ++t) b_c[t] = b_n[t];
  }
  // Pipeline epilogue: final K-chunk.
  #pragma unroll
  for (int t = 0; t < NT; ++t)
    acc[t] = __builtin_amdgcn_wmma_f32_16x16x4_f32(false, a_c, false, b_c[t],
                                                   (short)0, acc[t], false, false);

  // Store D + bias: row = m0 + vr + 8*half, col = n0 + 16*t + r.
  const int row_base = m0 + 8 * half;
  #pragma unroll
  for (int t = 0; t < NT; ++t) {
    const int col = n0 + 16 * t + r;
    const float bv = bias[col];
    #pragma unroll
    for (int vr = 0; vr < 8; ++vr) {
      C[(size_t)(row_base + vr) * N + col] = acc[t][vr] + bv;
    }
  }
}

// ---------------------------------------------------------------------------
// Kernel 4: sequential LSTM scan. Single wave32; lane handles channels
// j and j+32 of each 64-wide gate block; 5-step shfl_xor butterfly gives the
// scalar h. The gate rows for step t+1 are DMA'd into LDS with
// GLOBAL_LOAD_ASYNC_TO_LDS_B128 (ASYNCcnt) while step t computes, hiding
// the per-step memory latency of this serial recurrence.
// ---------------------------------------------------------------------------
__device__ __forceinline__ float sigf(float x) {
  return 1.0f / (1.0f + __expf(-x));
}
__device__ __forceinline__ float tanh_f(float x) { return tanhf(x); }

__global__ __launch_bounds__(32)
void lstm_scan(const float* __restrict__ Gx, const float* __restrict__ Gf,
               const float* __restrict__ Whh, const float* __restrict__ Whr,
               float* __restrict__ out) {
  const int lane = threadIdx.x;
  // Preload recurrent weights (constant over time) into registers.
  const float wi0 = Whh[lane],        wi1 = Whh[lane + 32];
  const float wf0 = Whh[64 + lane],   wf1 = Whh[96 + lane];
  const float wg0 = Whh[128 + lane],  wg1 = Whh[160 + lane];
  const float wo0 = Whh[192 + lane],  wo1 = Whh[224 + lane];
  const float wr0 = Whr[lane],        wr1 = Whr[lane + 32];

  float c0 = 0.f, c1 = 0.f, h = 0.f;

#if HAVE_ASYNC_LDS
  // Double-buffered gate rows: [stage][0:256)=Gx row, [256:512)=Gf row.
  __shared__ float sbuf[2][512];
  // Each lane DMAs 16B; two B128 ops cover one 1KB row (the immediate offset
  // applies to both the global and LDS side per ISA async addressing).
  auto issue = [&](int st, int t) {
    gvec_t gx = (gvec_t)(Gx + (size_t)t * 256 + lane * 4);
    gvec_t gf = (gvec_t)(Gf + (size_t)t * 256 + lane * 4);
    lvec_t lx = (lvec_t)&sbuf[st][lane * 4];
    lvec_t lf = (lvec_t)&sbuf[st][256 + lane * 4];
    __builtin_amdgcn_global_load_async_to_lds_b128(gx, lx, 0, 0);
    __builtin_amdgcn_global_load_async_to_lds_b128(gx, lx, 512, 0);
    __builtin_amdgcn_global_load_async_to_lds_b128(gf, lf, 0, 0);
    __builtin_amdgcn_global_load_async_to_lds_b128(gf, lf, 512, 0);
  };
  issue(0, 0);
  __builtin_amdgcn_s_wait_asynccnt(0);
#endif

  for (int t = 0; t < SS; ++t) {
#if HAVE_ASYNC_LDS
    if (t + 1 < SS) issue((t + 1) & 1, t + 1);   // prefetch next step's rows
    const float* g = &sbuf[t & 1][0];
    const float* q = &sbuf[t & 1][256];
#else
    const float* g = Gx + (size_t)t * 256;
    const float* q = Gf + (size_t)t * 256;
#endif
    // ---- main cell: gates = Gx[t] + Whh*h -------------------------------
    float i0 = g[lane]        + wi0 * h, i1 = g[lane + 32]  + wi1 * h;
    float f0 = g[lane + 64]   + wf0 * h, f1 = g[lane + 96]  + wf1 * h;
    float gg0 = g[lane + 128] + wg0 * h, gg1 = g[lane + 160] + wg1 * h;
    float o0 = g[lane + 192]  + wo0 * h, o1 = g[lane + 224] + wo1 * h;

    c0 = sigf(f0) * c0 + sigf(i0) * tanh_f(gg0);
    c1 = sigf(f1) * c1 + sigf(i1) * tanh_f(gg1);

    float part = wr0 * sigf(o0) * tanh_f(c0) + wr1 * sigf(o1) * tanh_f(c1);
    #pragma unroll
    for (int m = 16; m > 0; m >>= 1) part += __shfl_xor(part, m, 32);
    h = part;   // new scalar hidden state, identical in all lanes

    // ---- forecast cell (does not update the carry) ----------------------
    float pi0 = q[lane]        + wi0 * h, pi1 = q[lane + 32]  + wi1 * h;
    float pf0 = q[lane + 64]   + wf0 * h, pf1 = q[lane + 96]  + wf1 * h;
    float pg0 = q[lane + 128]  + wg0 * h, pg1 = q[lane + 160] + wg1 * h;
    float po0 = q[lane + 192]  + wo0 * h, po1 = q[lane + 224] + wo1 * h;

    float cf0 = sigf(pf0) * c0 + sigf(pi0) * tanh_f(pg0);
    float cf1 = sigf(pf1) * c1 + sigf(pi1) * tanh_f(pg1);

    float part2 = wr0 * sigf(po0) * tanh_f(cf0) + wr1 * sigf(po1) * tanh_f(cf1);
    #pragma unroll
    for (int m = 16; m > 0; m >>= 1) part2 += __shfl_xor(part2, m, 32);

    if (lane == 0) {
      #pragma unroll
      for (int b = 0; b < BB; ++b) {
        out[b * SS + t] = h;                 // progress
        out[BB * SS + b * SS + t] = part2;   // forecasted_progress
      }
    }
#if HAVE_ASYNC_LDS
    // Next iteration reads stage (t+1)&1 from LDS: ensure DMA landed.
    __builtin_amdgcn_s_wait_asynccnt(0);
#endif
  }
}

// ---------------------------------------------------------------------------
// Launch
// ---------------------------------------------------------------------------
extern "C" void kernel_launch(void* const* d_in, const int* in_sizes, int n_in,
                              void* d_out, int out_size, void* d_ws, size_t ws_size,
                              hipStream_t stream) {
  (void)in_sizes; (void)n_in; (void)out_size; (void)ws_size;

  const float* x   = (const float*)d_in[0];   // (B,S,D)
  const float* W1  = (const float*)d_in[1];   // (FH,D)
  const float* b1  = (const float*)d_in[2];   // (FH)
  const float* W2  = (const float*)d_in[3];   // (D,FH)
  const float* b2  = (const float*)d_in[4];   // (D)
  const float* Wih = (const float*)d_in[5];   // (4H,D)
  const float* Whh = (const float*)d_in[6];   // (4H,1)
  const float* bih = (const float*)d_in[7];   // (4H)
  const float* bhh = (const float*)d_in[8];   // (4H)
  const float* Whr = (const float*)d_in[9];   // (1,H)

  float* out = (float*)d_out;
  // d_out layout: progress[B*S] | forecasted[B*S] | fore[B*S*D]
  float* fore = out + 2 * BB * SS;

  // Workspace layout (floats): Weff[512*512] beff[512] bsum[256] Gx[S*256] Gf[S*256]
  float* ws   = (float*)d_ws;
  float* Weff = ws;
  float* beff = Weff + DD * DD;
  float* bsum = beff + DD;
  float* Gx   = bsum + 256;
  float* Gf   = Gx + (size_t)SS * 256;

  // 1) Collapse the linear MLP: Weff = W2 @ W1, beff = W2@b1 + b2
  fuse_weights<<<(DD * DD) / 256, 256, 0, stream>>>(W1, W2, Weff);
  prep_bias<<<1, 512, 0, stream>>>(W2, b1, b2, bih, bhh, beff, bsum);

  // 2) fore = flat @ Weff^T + beff   (M=16384, N=512, K=512), written to d_out
  {
    dim3 grid((BB * SS) / 64, DD / (16 * NT));
    wmma_gemm_bias<DD><<<grid, 128, 0, stream>>>(x, Weff, beff, fore, DD);
  }

  // 3) Gate pre-projections for the scan (Wih@inp + bih + bhh), hoisted:
  //    Gx = xs @ Wih^T + bsum   (xs = first S rows of flat = x[0])
  //    Gf = fore[:S] @ Wih^T + bsum
  {
    dim3 grid(SS / 64, 256 / (16 * NT));
    wmma_gemm_bias<DD><<<grid, 128, 0, stream>>>(x, Wih, bsum, Gx, 256);
    wmma_gemm_bias<DD><<<grid, 128, 0, stream>>>(fore, Wih, bsum, Gf, 256);
  }

  // 4) Sequential scan: one wave32, scalar-h recurrence with async-LDS
  //    double buffering of the gate rows.
  lstm_scan<<<1, 32, 0, stream>>>(Gx, Gf, Whh, Whr, out);
}